// FlairLinearChainCRF_48576080118511
// MI455X (gfx1250) — compile-verified
//
#include <hip/hip_runtime.h>
#include <hip/hip_bf16.h>
#include <math.h>

#define NTAG     48
#define NN       50          // NTAG + 2
#define NPAD     64          // padded tag dim for WMMA
#define SS       512
#define BB       1024
#define START_ID 48
#define STOP_ID  49
#define NEGV     (-1.0e12f)

typedef __attribute__((ext_vector_type(2))) float v2f;
typedef __attribute__((ext_vector_type(8))) float v8f;

#if defined(__HIP_DEVICE_COMPILE__) && __has_builtin(__builtin_amdgcn_exp2f)
#define RAW_EXP2(x) __builtin_amdgcn_exp2f(x)
#else
#define RAW_EXP2(x) exp2f(x)
#endif
#if defined(__HIP_DEVICE_COMPILE__) && __has_builtin(__builtin_amdgcn_logf)
#define RAW_LOG2(x) __builtin_amdgcn_logf(x)
#else
#define RAW_LOG2(x) log2f(x)
#endif

#define LOG2E_F 1.44269504088896340736f
#define LN2_F   0.69314718055994530942f

// ---------------------------------------------------------------------------
// prep: ET[k][j] = exp(trans[j][k]) padded to 64x64 (pad = 0),
//       tstop[j] = trans[STOP_ID][j] padded with NEG.
// ---------------------------------------------------------------------------
__global__ __launch_bounds__(256) void crf_prep_kernel(const float* __restrict__ trans,
                                                       float* __restrict__ ET,
                                                       float* __restrict__ tstop) {
    int i = blockIdx.x * blockDim.x + threadIdx.x;
    if (i < NPAD * NPAD) {
        int k = i >> 6;          // row of ET (the contraction index)
        int j = i & (NPAD - 1);  // col of ET (the output tag)
        float v = 0.0f;
        if (k < NN && j < NN) v = __expf(trans[j * NN + k]);
        ET[i] = v;
    }
    if (i < NPAD) tstop[i] = (i < NN) ? trans[STOP_ID * NN + i] : NEGV;
}

// ---------------------------------------------------------------------------
// forward algorithm: one block = 16 batch rows, 4 waves, each wave owns one
// 16-wide j-tile. Per timestep:
//   (prefetch emissions for t -- independent of alpha, hides HBM latency)
//   m[row]   = max_k alpha[row][k]         (shfl-reduced, 1 barrier)
//   P[row,k] = exp2((alpha - m) * log2e)
//   Q        = P(16x64) x ET(64x64)        via v_wmma_f32_16x16x4_f32
//   alpha    = mask(t < len, emit + m + log(Q), alpha)   (branchless)
// ---------------------------------------------------------------------------
__global__ __launch_bounds__(128) void crf_forward_kernel(const float* __restrict__ feats,
                                                          const int*   __restrict__ lens,
                                                          const float* __restrict__ ET,
                                                          const float* __restrict__ tstop,
                                                          float* __restrict__ fwd) {
    __shared__ float sET[NPAD * NPAD];    // 16 KB, B-operand source (k-major)
    __shared__ float sAlpha[16 * NPAD];   // 4 KB
    __shared__ float sP[16 * NPAD];       // 4 KB
    __shared__ float sMax[16];
    __shared__ float sStop[NPAD];
    __shared__ int   sLen[16];

    const int tid = threadIdx.x;
    const int b0  = blockIdx.x * 16;

    for (int i = tid; i < NPAD * NPAD; i += 128) sET[i] = ET[i];
    if (tid < 16) sLen[tid] = lens[b0 + tid];
    if (tid < NPAD) sStop[tid] = tstop[tid];
    for (int i = tid; i < 16 * NPAD; i += 128) {
        int k = i & (NPAD - 1);
        sAlpha[i] = (k == START_ID) ? 0.0f : NEGV;
    }
    __syncthreads();

    const int lane = tid & 31;
    const int wv   = tid >> 5;      // wave id == j-tile id (0..3)
    const int lo   = lane & 15;
    const int hi   = lane >> 4;

    const int col  = wv * 16 + lo;
    const int ccol = (col < NN) ? col : 0;      // clamped, branchless load addr

    // row-max assignment: 8 consecutive lanes (same wave) share one row
    const int mrow = tid >> 3;      // 0..15
    const int mseg = tid & 7;       // 0..7

    for (int t = 0; t < SS; ++t) {
        // --- prefetch emissions for this step (independent of alpha) -------
        float emitv[8];
#pragma unroll
        for (int r = 0; r < 8; ++r) {
            int row = r + 8 * hi;
            emitv[r] = feats[((b0 + row) * SS + t) * NN + ccol];
        }

        // --- row maxima: 8 lanes per row, shfl-xor reduce within wave ------
        {
            const float* ap = &sAlpha[mrow * NPAD + mseg * 8];
            float pm = ap[0];
#pragma unroll
            for (int i = 1; i < 8; ++i) pm = fmaxf(pm, ap[i]);
            pm = fmaxf(pm, __shfl_xor(pm, 1, 32));
            pm = fmaxf(pm, __shfl_xor(pm, 2, 32));
            pm = fmaxf(pm, __shfl_xor(pm, 4, 32));
            if (mseg == 0) sMax[mrow] = pm;
        }
        __syncthreads();

        // --- P = exp2((alpha - m) * log2e) ---------------------------------
#pragma unroll
        for (int i = 0; i < 8; ++i) {
            int idx = tid + i * 128;
            int m   = idx >> 6;
            sP[idx] = RAW_EXP2((sAlpha[idx] - sMax[m]) * LOG2E_F);
        }
        __syncthreads();

        // --- GEMM tile via fp32 WMMA ---------------------------------------
        // A (16x4 f32): lane lo = M, half hi selects K pair {0,1} vs {2,3}
        // B (4x16 f32): symmetric layout, ET is stored k-major so B=ET slice
        v8f acc = {};
#pragma unroll
        for (int k0 = 0; k0 < NPAD; k0 += 4) {
            v2f a, b;
            a.x = sP[lo * NPAD + k0 + 2 * hi];
            a.y = sP[lo * NPAD + k0 + 2 * hi + 1];
            b.x = sET[(k0 + 2 * hi) * NPAD + col];
            b.y = sET[(k0 + 2 * hi + 1) * NPAD + col];
            acc = __builtin_amdgcn_wmma_f32_16x16x4_f32(
                false, a, false, b, (short)0, acc, false, false);
        }

        // --- alpha update: lane holds D[r + 8*hi][lo] in acc[r] -------------
#pragma unroll
        for (int r = 0; r < 8; ++r) {
            int row = r + 8 * hi;
            float old = sAlpha[row * NPAD + col];
            float nv  = fmaxf(emitv[r] + sMax[row] + RAW_LOG2(acc[r]) * LN2_F, NEGV);
            nv = (col < NN) ? nv : NEGV;
            sAlpha[row * NPAD + col] = (t < sLen[row]) ? nv : old;
        }
        __syncthreads();
    }

    // --- terminal logsumexp(alpha + trans[STOP]) per row -------------------
    if (tid < 16) {
        int row = tid;
        float rmax = NEGV * 4.0f;
        for (int j = 0; j < NPAD; ++j) {
            float v = sAlpha[row * NPAD + j] + sStop[j];
            rmax = fmaxf(rmax, v);
        }
        float s = 0.0f;
        for (int j = 0; j < NPAD; ++j) {
            float v = sAlpha[row * NPAD + j] + sStop[j];
            s += RAW_EXP2((v - rmax) * LOG2E_F);
        }
        fwd[b0 + row] = rmax + RAW_LOG2(s) * LN2_F;
    }
}

// ---------------------------------------------------------------------------
// gold score: one block per batch row
// ---------------------------------------------------------------------------
__global__ __launch_bounds__(128) void crf_gold_kernel(const float* __restrict__ feats,
                                                       const int*   __restrict__ tags,
                                                       const int*   __restrict__ lens,
                                                       const float* __restrict__ trans,
                                                       float* __restrict__ gold) {
    const int b   = blockIdx.x;
    const int tid = threadIdx.x;
    const int len = lens[b];
    float acc = 0.0f;
    for (int t = tid; t <= len; t += 128) {     // transitions over t in [0, len]
        int prev = (t == 0) ? START_ID : tags[b * SS + t - 1];
        int cur  = (t < len) ? tags[b * SS + t] : STOP_ID;
        acc += trans[cur * NN + prev];
        if (t < len) acc += feats[(b * SS + t) * NN + cur];  // emission for t < len
    }
    __shared__ float red[128];
    red[tid] = acc;
    __syncthreads();
    for (int s = 64; s > 0; s >>= 1) {
        if (tid < s) red[tid] += red[tid + s];
        __syncthreads();
    }
    if (tid == 0) gold[b] = red[0];
}

// ---------------------------------------------------------------------------
// final: mean(gold - forward)
// ---------------------------------------------------------------------------
__global__ __launch_bounds__(256) void crf_final_kernel(const float* __restrict__ gold,
                                                        const float* __restrict__ fwd,
                                                        float* __restrict__ out) {
    __shared__ float red[256];
    const int tid = threadIdx.x;
    float acc = 0.0f;
    for (int i = tid; i < BB; i += 256) acc += gold[i] - fwd[i];
    red[tid] = acc;
    __syncthreads();
    for (int s = 128; s > 0; s >>= 1) {
        if (tid < s) red[tid] += red[tid + s];
        __syncthreads();
    }
    if (tid == 0) out[0] = red[0] / (float)BB;
}

// ---------------------------------------------------------------------------
extern "C" void kernel_launch(void* const* d_in, const int* in_sizes, int n_in,
                              void* d_out, int out_size, void* d_ws, size_t ws_size,
                              hipStream_t stream) {
    const float* feats = (const float*)d_in[0];   // [B,S,N] f32
    const int*   tags  = (const int*)  d_in[1];   // [B,S]   i32
    const int*   lens  = (const int*)  d_in[2];   // [B]     i32
    const float* trans = (const float*)d_in[3];   // [N,N]   f32
    float* out = (float*)d_out;

    float* ws    = (float*)d_ws;
    float* ET    = ws;                       // 64*64
    float* tstop = ET + NPAD * NPAD;         // 64
    float* fwd   = tstop + NPAD;             // 1024
    float* gold  = fwd + BB;                 // 1024

    crf_prep_kernel<<<(NPAD * NPAD + 255) / 256, 256, 0, stream>>>(trans, ET, tstop);
    crf_forward_kernel<<<BB / 16, 128, 0, stream>>>(feats, lens, ET, tstop, fwd);
    crf_gold_kernel<<<BB, 128, 0, stream>>>(feats, tags, lens, trans, gold);
    crf_final_kernel<<<1, 256, 0, stream>>>(gold, fwd, out);
}